// AttnDecoderRNN_20770461843758
// MI455X (gfx1250) — compile-verified
//
#include <hip/hip_runtime.h>
#include <math.h>

#define H       256
#define L       256
#define NLAYERS 4
#define VOCAB   50257

typedef float v2f __attribute__((ext_vector_type(2)));
typedef float v8f __attribute__((ext_vector_type(8)));

// ---------- wave32 reduction ----------
__device__ __forceinline__ float waveReduce(float v) {
    #pragma unroll
    for (int off = 16; off > 0; off >>= 1)
        v += __shfl_xor(v, off, 32);
    return v;
}

// ---------------------------------------------------------------------------
// 1. embedding lookup: xe[h] = emb[token, h]
// ---------------------------------------------------------------------------
__global__ __launch_bounds__(256) void kEmbed(const int* __restrict__ token,
                                              const float* __restrict__ emb,
                                              float* __restrict__ xe) {
    int tok = token[0];
    xe[threadIdx.x] = emb[(size_t)tok * H + threadIdx.x];
}

// ---------------------------------------------------------------------------
// 2. attention scores: scores[l] = dot([emb,hid0], attn_w[l]) + attn_b[l]
//    grid 8 x 256 (wave-per-row, coalesced float4)
// ---------------------------------------------------------------------------
__global__ __launch_bounds__(256) void kAttnScores(const float* __restrict__ xe,
                                                   const float* __restrict__ hid0,
                                                   const float* __restrict__ attn_w,
                                                   const float* __restrict__ attn_b,
                                                   float* __restrict__ scores) {
    __shared__ float cat[2 * H];
    int tid = threadIdx.x, lane = tid & 31, wave = tid >> 5;
    cat[tid]     = xe[tid];
    cat[H + tid] = hid0[tid];
    __syncthreads();
    int row0 = blockIdx.x * 32;
    for (int rr = wave; rr < 32; rr += 8) {
        int row = row0 + rr;
        const float* w = attn_w + (size_t)row * (2 * H);
        float acc = 0.f;
        for (int c = lane * 4; c < 2 * H; c += 128) {
            float4 w4 = *(const float4*)(w + c);
            acc += w4.x * cat[c] + w4.y * cat[c + 1] + w4.z * cat[c + 2] + w4.w * cat[c + 3];
        }
        acc = waveReduce(acc);
        if (lane == 0) scores[row] = acc + attn_b[row];
    }
}

// ---------------------------------------------------------------------------
// 3. softmax over 256 scores -> aw (ws) and attn_weights output
// ---------------------------------------------------------------------------
__global__ __launch_bounds__(256) void kSoftmax256(const float* __restrict__ scores,
                                                   float* __restrict__ aw,
                                                   float* __restrict__ out_attn) {
    __shared__ float red[256];
    int tid = threadIdx.x;
    float s = scores[tid];
    red[tid] = s; __syncthreads();
    for (int k = 128; k > 0; k >>= 1) {
        if (tid < k) red[tid] = fmaxf(red[tid], red[tid + k]);
        __syncthreads();
    }
    float mx = red[0]; __syncthreads();
    float e = expf(s - mx);
    red[tid] = e; __syncthreads();
    for (int k = 128; k > 0; k >>= 1) {
        if (tid < k) red[tid] += red[tid + k];
        __syncthreads();
    }
    float w = e / red[0];
    aw[tid] = w;
    out_attn[tid] = w;
}

// ---------------------------------------------------------------------------
// 4. attn_applied partials: block b covers l in [32b,32b+32)
//    part[b][h] = sum_l aw[l]*enc[l][h]   (coalesced over h)
// ---------------------------------------------------------------------------
__global__ __launch_bounds__(256) void kAttnApply(const float* __restrict__ aw,
                                                  const float* __restrict__ enc,
                                                  float* __restrict__ part) {
    __shared__ float w[32];
    int tid = threadIdx.x;
    int l0 = blockIdx.x * 32;
    if (tid < 32) w[tid] = aw[l0 + tid];
    __syncthreads();
    float acc = 0.f;
    #pragma unroll
    for (int i = 0; i < 32; ++i)
        acc += w[i] * enc[(size_t)(l0 + i) * H + tid];
    part[blockIdx.x * H + tid] = acc;
}

// ---------------------------------------------------------------------------
// 5. combine: x[h] = relu(dot([emb, attn_applied], comb_w[h]) + comb_b[h])
//    grid 8 x 256; each block redundantly reduces the 8 partials (tiny)
// ---------------------------------------------------------------------------
__global__ __launch_bounds__(256) void kCombine(const float* __restrict__ xe,
                                                const float* __restrict__ part,
                                                const float* __restrict__ comb_w,
                                                const float* __restrict__ comb_b,
                                                float* __restrict__ xout) {
    __shared__ float cat[2 * H];
    int tid = threadIdx.x, lane = tid & 31, wave = tid >> 5;
    float aa = 0.f;
    #pragma unroll
    for (int b = 0; b < 8; ++b) aa += part[b * H + tid];
    cat[tid]     = xe[tid];
    cat[H + tid] = aa;
    __syncthreads();
    int row0 = blockIdx.x * 32;
    for (int rr = wave; rr < 32; rr += 8) {
        int row = row0 + rr;
        const float* w = comb_w + (size_t)row * (2 * H);
        float acc = 0.f;
        for (int c = lane * 4; c < 2 * H; c += 128) {
            float4 w4 = *(const float4*)(w + c);
            acc += w4.x * cat[c] + w4.y * cat[c + 1] + w4.z * cat[c + 2] + w4.w * cat[c + 3];
        }
        acc = waveReduce(acc);
        if (lane == 0) {
            float v = acc + comb_b[row];
            xout[row] = v > 0.f ? v : 0.f;
        }
    }
}

// ---------------------------------------------------------------------------
// 6a. GRU mat-vecs for one layer: gi[0:768], gh[0:768]; grid 12 x 256
// ---------------------------------------------------------------------------
__global__ __launch_bounds__(256) void kGruMat(const float* __restrict__ x,
                                               const float* __restrict__ hprev,
                                               const float* __restrict__ wih,
                                               const float* __restrict__ whh,
                                               const float* __restrict__ bih,
                                               const float* __restrict__ bhh,
                                               float* __restrict__ gi,
                                               float* __restrict__ gh) {
    __shared__ float xs[H], hs[H];
    int tid = threadIdx.x, lane = tid & 31, wave = tid >> 5;
    xs[tid] = x[tid];
    hs[tid] = hprev[tid];
    __syncthreads();
    int rowStart = blockIdx.x * 128;
    for (int rr = wave; rr < 128; rr += 8) {
        int row = rowStart + rr;                 // 0..1535
        bool isGi = row < 3 * H;
        int r = isGi ? row : row - 3 * H;
        const float* w = (isGi ? wih : whh) + (size_t)r * H;
        const float* v = isGi ? xs : hs;
        float acc = 0.f;
        for (int c = lane * 4; c < H; c += 128) {
            float4 w4 = *(const float4*)(w + c);
            acc += w4.x * v[c] + w4.y * v[c + 1] + w4.z * v[c + 2] + w4.w * v[c + 3];
        }
        acc = waveReduce(acc);
        if (lane == 0) {
            if (isGi) gi[r] = acc + bih[r];
            else      gh[r] = acc + bhh[r];
        }
    }
}

// ---------------------------------------------------------------------------
// 6b. GRU gate elementwise (r,z,n ordering, PyTorch convention)
// ---------------------------------------------------------------------------
__global__ __launch_bounds__(256) void kGruGate(const float* __restrict__ gi,
                                                const float* __restrict__ gh,
                                                const float* __restrict__ hprev,
                                                float* __restrict__ hout,
                                                float* __restrict__ xout) {
    int j = threadIdx.x;
    float r = 1.f / (1.f + expf(-(gi[j] + gh[j])));
    float z = 1.f / (1.f + expf(-(gi[H + j] + gh[H + j])));
    float n = tanhf(gi[2 * H + j] + r * gh[2 * H + j]);
    float h = hprev[j];
    float hn = (1.f - z) * n + z * h;
    hout[j] = hn;
    xout[j] = hn;
}

// ---------------------------------------------------------------------------
// 7. output projection via WMMA f32 16x16x4.
//    Block = 64 threads (2 waves) handles 32 vocab rows.
//    Stage 32x256 f32 tile into LDS (coalesced b128), padded stride 260
//    (bank-conflict-free for the paired b64 fragment reads), then 64
//    v_wmma_f32_16x16x4_f32 per wave with B = x broadcast across N.
// ---------------------------------------------------------------------------
#define VTILE 32
#define WSTRIDE 260

__global__ __launch_bounds__(64) void kLogits(const float* __restrict__ x,
                                              const float* __restrict__ out_w,
                                              const float* __restrict__ out_b,
                                              float* __restrict__ logits) {
    __shared__ float xs[H];
    __shared__ float wt[VTILE * WSTRIDE];
    const int tid  = threadIdx.x;      // 0..63
    const int lane = tid & 31;
    const int wave = tid >> 5;         // 0..1
    const int row0 = blockIdx.x * VTILE;

    for (int i = tid; i < H; i += 64) xs[i] = x[i];

    // stage weight tile: 32 rows x 64 float4, fully coalesced global reads
    for (int i = tid; i < VTILE * (H / 4); i += 64) {
        int r  = i >> 6;        // row in tile
        int c4 = i & 63;        // float4 column
        int gr = row0 + r;
        float4 w4 = make_float4(0.f, 0.f, 0.f, 0.f);
        if (gr < VOCAB) w4 = *(const float4*)(out_w + (size_t)gr * H + c4 * 4);
        float* dst = &wt[r * WSTRIDE + c4 * 4];
        dst[0] = w4.x; dst[1] = w4.y; dst[2] = w4.z; dst[3] = w4.w;
    }
    __syncthreads();

    // A fragment: lane L<16 holds A[M=L][K=k..k+1]; lane>=16 holds A[M=L-16][K=k+2..k+3]
    const int m    = lane & 15;
    const int koff = (lane >> 4) * 2;
    const float* arow = &wt[(wave * 16 + m) * WSTRIDE];

    v8f c = {0.f, 0.f, 0.f, 0.f, 0.f, 0.f, 0.f, 0.f};
    #pragma unroll 8
    for (int k = 0; k < H; k += 4) {
        v2f a = *(const v2f*)(arow + k + koff);     // LDS, conflict-free
        v2f b = *(const v2f*)(&xs[k + koff]);       // LDS broadcast of x
        c = __builtin_amdgcn_wmma_f32_16x16x4_f32(false, a, false, b,
                                                  (short)0, c, false, false);
    }

    // D layout: vgpr v, lane 0 -> M=v (N=0); lane 16 -> M=v+8 (N=0)
    if ((lane & 15) == 0) {
        int base = row0 + wave * 16 + (lane >> 4) * 8;
        #pragma unroll
        for (int i = 0; i < 8; ++i) {
            int r = base + i;
            if (r < VOCAB) logits[r] = c[i] + out_b[r];
        }
    }
}

// ---------------------------------------------------------------------------
// 8. logZ = max + log(sum exp(logit - max)) over VOCAB (single 1024-thr block)
// ---------------------------------------------------------------------------
__global__ __launch_bounds__(1024) void kLogZ(const float* __restrict__ logits,
                                              float* __restrict__ logZ) {
    __shared__ float red[1024];
    int tid = threadIdx.x;
    float mx = -INFINITY;
    for (int i = tid; i < VOCAB; i += 1024) mx = fmaxf(mx, logits[i]);
    red[tid] = mx; __syncthreads();
    for (int s = 512; s > 0; s >>= 1) {
        if (tid < s) red[tid] = fmaxf(red[tid], red[tid + s]);
        __syncthreads();
    }
    float m = red[0]; __syncthreads();
    float sum = 0.f;
    for (int i = tid; i < VOCAB; i += 1024) sum += expf(logits[i] - m);
    red[tid] = sum; __syncthreads();
    for (int s = 512; s > 0; s >>= 1) {
        if (tid < s) red[tid] += red[tid + s];
        __syncthreads();
    }
    if (tid == 0) *logZ = m + logf(red[0]);
}

// ---------------------------------------------------------------------------
// 9. logp[v] = logits[v] - logZ  (in place in d_out)
// ---------------------------------------------------------------------------
__global__ __launch_bounds__(256) void kLogSoftmax(float* __restrict__ logits,
                                                   const float* __restrict__ logZ) {
    int v = blockIdx.x * 256 + threadIdx.x;
    if (v < VOCAB) logits[v] = logits[v] - logZ[0];
}

// ---------------------------------------------------------------------------
extern "C" void kernel_launch(void* const* d_in, const int* in_sizes, int n_in,
                              void* d_out, int out_size, void* d_ws, size_t ws_size,
                              hipStream_t stream) {
    const int*   token  = (const int*)  d_in[0];
    const float* hidden = (const float*)d_in[1];   // [4,1,256]
    const float* enc    = (const float*)d_in[2];   // [256,256]
    const float* emb    = (const float*)d_in[3];   // [VOCAB,256]
    const float* attn_w = (const float*)d_in[4];   // [256,512]
    const float* attn_b = (const float*)d_in[5];   // [256]
    const float* comb_w = (const float*)d_in[6];   // [256,512]
    const float* comb_b = (const float*)d_in[7];   // [256]
    const float* gwih   = (const float*)d_in[8];   // [4,768,256]
    const float* gwhh   = (const float*)d_in[9];   // [4,768,256]
    const float* gbih   = (const float*)d_in[10];  // [4,768]
    const float* gbhh   = (const float*)d_in[11];  // [4,768]
    const float* out_w  = (const float*)d_in[12];  // [VOCAB,256]
    const float* out_b  = (const float*)d_in[13];  // [VOCAB]

    float* out = (float*)d_out;
    float* out_logp = out;                 // [VOCAB]
    float* out_hid  = out + VOCAB;         // [4*256]
    float* out_attn = out + VOCAB + 4 * H; // [256]

    // workspace layout (floats)
    float* ws      = (float*)d_ws;
    float* ws_emb  = ws;            // 256
    float* ws_sc   = ws + 256;      // 256
    float* ws_aw   = ws + 512;      // 256
    float* ws_part = ws + 768;      // 8*256
    float* ws_x    = ws + 2816;     // 256
    float* ws_gi   = ws + 3072;     // 768
    float* ws_gh   = ws + 3840;     // 768
    float* ws_logz = ws + 4608;     // 1

    kEmbed<<<1, 256, 0, stream>>>(token, emb, ws_emb);
    kAttnScores<<<8, 256, 0, stream>>>(ws_emb, hidden /*layer0*/, attn_w, attn_b, ws_sc);
    kSoftmax256<<<1, 256, 0, stream>>>(ws_sc, ws_aw, out_attn);
    kAttnApply<<<8, 256, 0, stream>>>(ws_aw, enc, ws_part);
    kCombine<<<8, 256, 0, stream>>>(ws_emb, ws_part, comb_w, comb_b, ws_x);

    for (int l = 0; l < NLAYERS; ++l) {
        kGruMat<<<12, 256, 0, stream>>>(ws_x, hidden + (size_t)l * H,
                                        gwih + (size_t)l * 3 * H * H,
                                        gwhh + (size_t)l * 3 * H * H,
                                        gbih + (size_t)l * 3 * H,
                                        gbhh + (size_t)l * 3 * H,
                                        ws_gi, ws_gh);
        kGruGate<<<1, 256, 0, stream>>>(ws_gi, ws_gh, hidden + (size_t)l * H,
                                        out_hid + (size_t)l * H, ws_x);
    }

    int nTiles = (VOCAB + VTILE - 1) / VTILE;   // 1571
    kLogits<<<nTiles, 64, 0, stream>>>(ws_x, out_w, out_b, out_logp);
    kLogZ<<<1, 1024, 0, stream>>>(out_logp, ws_logz);
    kLogSoftmax<<<(VOCAB + 255) / 256, 256, 0, stream>>>(out_logp, ws_logz);
}